// RingBuffer_73160472920634
// MI455X (gfx1250) — compile-verified
//
#include <hip/hip_runtime.h>
#include <stdint.h>

// Problem constants (fixed by the reference setup).
#define C_CH     64
#define BUF_LEN  2097152u          // 2^21
#define BUF_MASK (BUF_LEN - 1u)
#define NSAMP    1048576           // 2^20
#define DELAY    4096
#define THREADS  256
#define UNROLL   4                 // float4s per thread in the bulk copy

typedef __attribute__((address_space(3))) void lds_void_t;

// 32-bit LDS byte address of a __shared__ object (what the async VFLAT
// instructions expect in their LDS-address VGPR).
__device__ __forceinline__ uint32_t lds_addr_of(const void* p) {
    return (uint32_t)(uintptr_t)(lds_void_t*)p;
}

// The 24-bit signed IOFFSET of the async ops is added to BOTH the LDS address
// and the global address (CDNA5 ISA), which matches our
// +16*u stride on both sides -> one address tuple per direction, offsets as
// immediates (must be string literals, hence macros).
#define ASYNC_LOAD_B128(ofs)                                                   \
    asm volatile("global_load_async_to_lds_b128 %0, %1, off offset:" #ofs      \
                 " th:TH_LOAD_NT" :: "v"(lds), "v"(src) : "memory")
#define ASYNC_STORE_B128(ofs)                                                  \
    asm volatile("global_store_async_from_lds_b128 %0, %1, off offset:" #ofs   \
                 " th:TH_STORE_NT" :: "v"(dst), "v"(lds) : "memory")

// ---------------------------------------------------------------------------
// Bulk: out[c, DELAY + k] = samples[c, k] for k in [0, NSAMP-DELAY)
// Streaming copy routed through the CDNA5 async copy engine:
// global_load_async_to_lds_b128 -> s_wait_asynccnt -> store_async_from_lds.
// No VGPR data staging; NT hints keep the touch-once streams out of L2.
// grid = (255, 64), block = 256 threads (8 wave32s). Exact division:
// (NSAMP - DELAY)/4 = 261120 float4 per channel = 255 * 256 * UNROLL.
// ---------------------------------------------------------------------------
__global__ void RingBuffer_bulk_async(const float4* __restrict__ samples4,
                                      float4* __restrict__ out4) {
    __shared__ float4 stage[THREADS * UNROLL];   // 16 KB, one slot per lane

    const int tid   = threadIdx.x;
    const int c     = blockIdx.y;
    const int chan4 = NSAMP / 4;                               // float4 per channel
    const int k4    = (blockIdx.x * THREADS + tid) * UNROLL;   // float4 idx in channel

    const float4* src = samples4 + (size_t)c * chan4 + k4;
    float4*       dst = out4     + (size_t)c * chan4 + (DELAY / 4) + k4;

    const uint32_t lds = lds_addr_of(&stage[tid * UNROLL]);

    // 4 async 16B/lane loads into this lane's private LDS slot (ASYNCcnt=4).
    ASYNC_LOAD_B128(0);
    ASYNC_LOAD_B128(16);
    ASYNC_LOAD_B128(32);
    ASYNC_LOAD_B128(48);
    // Wait for this wave's async loads to land in LDS before streaming out.
    asm volatile("s_wait_asynccnt 0x0" ::: "memory");
    ASYNC_STORE_B128(0);
    ASYNC_STORE_B128(16);
    ASYNC_STORE_B128(32);
    ASYNC_STORE_B128(48);
    // S_ENDPGM performs an implicit wait-idle, draining outstanding stores.
}

// ---------------------------------------------------------------------------
// Head: out[c, j] = buffer[c, (wi - DELAY + j) & mask] for j in [0, DELAY)
// Only 1 MB of traffic; simple coalesced scalar kernel. Reads wi on-device
// (low 32 bits are correct for int32 or little-endian int64 since wi < 2^21).
// ---------------------------------------------------------------------------
__global__ void RingBuffer_head_wrap(const float* __restrict__ buffer,
                                     const unsigned int* __restrict__ wi_ptr,
                                     float* __restrict__ out) {
    const int t = blockIdx.x * blockDim.x + threadIdx.x;  // 0 .. C*DELAY-1
    const int c = t >> 12;                                // / DELAY (=4096)
    const int j = t & (DELAY - 1);

    const unsigned int wi  = wi_ptr[0] & BUF_MASK;
    const unsigned int idx = (wi + BUF_LEN - DELAY + (unsigned)j) & BUF_MASK;

    out[(size_t)c * NSAMP + j] = buffer[(size_t)c * BUF_LEN + idx];
}

extern "C" void kernel_launch(void* const* d_in, const int* in_sizes, int n_in,
                              void* d_out, int out_size, void* d_ws, size_t ws_size,
                              hipStream_t stream) {
    const float*        buffer  = (const float*)d_in[0];
    const float*        samples = (const float*)d_in[1];
    const unsigned int* wi      = (const unsigned int*)d_in[2];
    float*              out     = (float*)d_out;

    // Bulk shifted copy: (NSAMP-DELAY)/4 float4 per channel / (THREADS*UNROLL)
    // = 255 blocks per channel, 64 channels.
    dim3 bulkGrid((NSAMP / 4 - DELAY / 4) / (THREADS * UNROLL), C_CH);
    RingBuffer_bulk_async<<<bulkGrid, THREADS, 0, stream>>>(
        (const float4*)samples, (float4*)out);

    // Head gather: C*DELAY = 262144 elements.
    RingBuffer_head_wrap<<<(C_CH * DELAY) / THREADS, THREADS, 0, stream>>>(
        buffer, wi, out);
}